// SubGraph_37915971289119
// MI455X (gfx1250) — compile-verified
//
#include <hip/hip_runtime.h>

// ---------------------------------------------------------------------------
// Problem constants (from the reference)
// ---------------------------------------------------------------------------
#define N_ROWS   1048576
#define D_IN     64
#define D_H      64
#define N_SEG    65536
#define LN_EPS   1e-5f

// LDS stride for transposed W: 66 floats => lane col*66 hits distinct banks
// (bank = (2*col + k) % 64), and stays 8-byte aligned (66*4 = 264 bytes).
#define WT_STRIDE 66

typedef float v2f __attribute__((ext_vector_type(2)));
typedef float v8f __attribute__((ext_vector_type(8)));

// ---------------------------------------------------------------------------
// Kernel 0: zero-init the segment-max table (d_ws). All fX >= 0 after ReLU and
// every gathered segment is non-empty, so identity 0.0f (bits 0) is exact.
// ---------------------------------------------------------------------------
__global__ __launch_bounds__(256) void init_segmax(int* __restrict__ p)
{
    int i = blockIdx.x * 256 + threadIdx.x;          // one int4 per thread
    ((int4*)p)[i] = make_int4(0, 0, 0, 0);           // N_SEG*D_H divisible by 4
}

// ---------------------------------------------------------------------------
// Kernel 1: fused  h = X@W + b  ->  LayerNorm -> gamma/beta -> ReLU
//           -> store fX to out[:, 0:64]  -> atomic scatter-max into segmax
//
// Wave mapping (wave32, V_WMMA_F32_16X16X4_F32):
//   - wave owns rows [r0, r0+16) of X; 4 accumulator tiles cover cols 0..63
//   - A (16x4 f32):  lane L<16 holds row L,  VGPR0/1 = K pair {4t, 4t+1}
//                    lane L>=16 holds row L-16, K pair {4t+2, 4t+3}
//   - B (4x16 f32):  VGPR v, lane-half hi -> K = 4t + 2*hi + v, N = lane&15
//     -> contiguous pair WT[col][4t+2*hi .. +1] from transposed-W LDS
//   - C/D (16x16):   VGPR r, lanes 0-15 -> row r, lanes 16-31 -> row r+8
// ---------------------------------------------------------------------------
__global__ __launch_bounds__(256) void fused_gemm_ln_relu_scatter(
    const float* __restrict__ X,     // [N_ROWS, 64]
    const int*   __restrict__ idx,   // [N_ROWS]
    const float* __restrict__ W,     // [64, 64] row-major (k, n)
    const float* __restrict__ bvec,  // [64]
    const float* __restrict__ gamma, // [64]
    const float* __restrict__ beta,  // [64]
    float*       __restrict__ out,   // [N_ROWS, 128]
    int*         __restrict__ segmax)// [N_SEG, 64] float bits
{
    __shared__ float WT[64 * WT_STRIDE];             // W transposed: WT[n][k]

    const int tid = threadIdx.x;

    // Stage W^T into LDS: coalesced global read, scattered LDS write.
    #pragma unroll
    for (int j = tid; j < 64 * 64; j += 256) {
        int k = j >> 6;
        int c = j & 63;
        WT[c * WT_STRIDE + k] = W[j];
    }
    __syncthreads();

    const int wave = tid >> 5;
    const int lane = tid & 31;
    const int hi   = lane >> 4;        // 0: lanes 0-15, 1: lanes 16-31
    const int lcol = lane & 15;        // N within a 16-wide tile

    const long r0 = (long)blockIdx.x * 128 + (long)wave * 16;

    // Per-lane column constants for the 4 N-tiles.
    float bb[4], gg[4], be[4];
    #pragma unroll
    for (int n = 0; n < 4; ++n) {
        int col = n * 16 + lcol;
        bb[n] = bvec[col];
        gg[n] = gamma[col];
        be[n] = beta[col];
    }

    // ---- GEMM: 16 chained K=4 FP32 WMMAs per tile ----
    v8f acc[4] = {v8f{}, v8f{}, v8f{}, v8f{}};
    const float* Arow = X + (r0 + lcol) * D_IN + 2 * hi;

    #pragma unroll
    for (int t = 0; t < 16; ++t) {
        v2f a = *(const v2f*)(Arow + 4 * t);                 // global_load_b64
        #pragma unroll
        for (int n = 0; n < 4; ++n) {
            int col = n * 16 + lcol;
            v2f bf = *(const v2f*)&WT[col * WT_STRIDE + 4 * t + 2 * hi]; // ds_load_b64
            acc[n] = __builtin_amdgcn_wmma_f32_16x16x4_f32(
                false, a, false, bf, (short)0, acc[n], false, false);
        }
    }

    // ---- bias ----
    #pragma unroll
    for (int n = 0; n < 4; ++n)
        #pragma unroll
        for (int r = 0; r < 8; ++r)
            acc[n][r] += bb[n];

    // ---- LayerNorm + affine + ReLU + store + scatter-max, row by row ----
    #pragma unroll
    for (int r = 0; r < 8; ++r) {
        float s = acc[0][r] + acc[1][r] + acc[2][r] + acc[3][r];
        float q = acc[0][r] * acc[0][r] + acc[1][r] * acc[1][r] +
                  acc[2][r] * acc[2][r] + acc[3][r] * acc[3][r];
        // Reduce across the 16-lane half that owns this row (masks < 16 keep
        // lanes 0-15 and 16-31 independent — exactly the C-layout row split).
        #pragma unroll
        for (int m = 1; m <= 8; m <<= 1) {
            s += __shfl_xor(s, m, 32);
            q += __shfl_xor(q, m, 32);
        }
        const float mean = s * (1.0f / 64.0f);
        const float var  = q * (1.0f / 64.0f) - mean * mean;
        const float rs   = rsqrtf(var + LN_EPS);

        const long grow = r0 + r + 8 * hi;       // global row for this lane
        const int  seg  = idx[grow];
        float* orow = out + grow * 128;
        int*   srow = segmax + (long)seg * D_H;

        #pragma unroll
        for (int n = 0; n < 4; ++n) {
            float y = (acc[n][r] - mean) * rs;
            y = fmaxf(y * gg[n] + be[n], 0.0f);
            orow[n * 16 + lcol] = y;
            // non-negative floats: int-bit order == float order
            atomicMax(&srow[n * 16 + lcol], __float_as_int(y));
        }
    }
}

// ---------------------------------------------------------------------------
// Kernel 2: out[:, 64:128] = segmax[idx[row]]  (float4 per thread; the 16 MB
// table is L2-resident on MI455X's 192 MB L2)
// ---------------------------------------------------------------------------
__global__ __launch_bounds__(256) void gather_concat(
    const int*   __restrict__ idx,
    const float* __restrict__ segmax,
    float*       __restrict__ out)
{
    long t   = (long)blockIdx.x * 256 + threadIdx.x;
    long row = t >> 4;                    // 16 threads (float4 each) per row
    int  c4  = (int)(t & 15);
    int  seg = idx[row];
    float4 v = ((const float4*)(segmax + (long)seg * D_H))[c4];
    ((float4*)(out + row * 128 + D_H))[c4] = v;
}

// ---------------------------------------------------------------------------
// Launch
// ---------------------------------------------------------------------------
extern "C" void kernel_launch(void* const* d_in, const int* in_sizes, int n_in,
                              void* d_out, int out_size, void* d_ws, size_t ws_size,
                              hipStream_t stream)
{
    const float* X     = (const float*)d_in[0];
    const int*   idx   = (const int*)  d_in[1];
    const float* W     = (const float*)d_in[2];
    const float* bvec  = (const float*)d_in[3];
    const float* gamma = (const float*)d_in[4];
    const float* beta  = (const float*)d_in[5];
    float* out   = (float*)d_out;
    int*   smax  = (int*)d_ws;                    // N_SEG*D_H floats = 16 MB

    // 0) init segment-max identity (harness poisons d_ws; re-init every call)
    init_segmax<<<(N_SEG * D_H / 4) / 256, 256, 0, stream>>>(smax);

    // 1) fused GEMM + LN + ReLU + scatter-max; 128 rows per 256-thread block
    fused_gemm_ln_relu_scatter<<<N_ROWS / 128, 256, 0, stream>>>(
        X, idx, W, bvec, gamma, beta, out, smax);

    // 2) gather segment maxima into second half of output
    gather_concat<<<(N_ROWS * 16) / 256, 256, 0, stream>>>(
        idx, (const float*)smax, out);
}